// VariableSortedHistoryPooling_17145509445965
// MI455X (gfx1250) — compile-verified
//
#include <hip/hip_runtime.h>
#include <stdint.h>

// ---------------------------------------------------------------------------
// VariableSortedHistoryPooling: out[u,d] = (1/max(len_u,1)) * sum_{e in seg_u}
//                               emb_weight[event_indices[e], d]
// Memory-bound gather+segment-mean. No matmul structure -> no WMMA.
// gfx1250 paths used: global_load_async_to_lds_b128 (ASYNCcnt) for index
// staging, global_prefetch_b8 on the gather stream, b128 vector gathers.
// ---------------------------------------------------------------------------

#define BLOCK_USERS 16     // 256 threads = 8 waves; half-wave (16 lanes) per user
#define IDX_CAP     2048   // LDS int slots for staged indices (8 KB)

#if defined(__has_builtin)
#if __has_builtin(__builtin_amdgcn_global_load_async_to_lds_b128) && \
    __has_builtin(__builtin_amdgcn_s_wait_asynccnt)
#define HAVE_ASYNC_LDS 1
#endif
#endif
#ifndef HAVE_ASYNC_LDS
#define HAVE_ASYNC_LDS 0
#endif

#if HAVE_ASYNC_LDS
// Builtin signature (from hipcc diagnostic): param0 = int4* in addrspace(1)
// ("__device__"), param1 = int4* in addrspace(3), then imm offset, imm cpol.
typedef int v4i_t __attribute__((vector_size(4 * sizeof(int))));
typedef __attribute__((address_space(1))) v4i_t as1_v4i;
typedef __attribute__((address_space(3))) v4i_t as3_v4i;
#endif

__global__ __launch_bounds__(256) void
VariableSortedHistoryPooling_kernel(const int* __restrict__ ev,
                                    const int* __restrict__ offs,
                                    const float* __restrict__ emb,
                                    float* __restrict__ out,
                                    int n_users)
{
    __shared__ int s_idx[IDX_CAP];

    const int tid = threadIdx.x;
    const int u0  = blockIdx.x * BLOCK_USERS;
    int u1 = u0 + BLOCK_USERS;
    if (u1 > n_users) u1 = n_users;

    const int  base_ev = offs[u0];
    const int  total   = offs[u1] - base_ev;
    const bool staged  = (total > 0) && (total <= IDX_CAP) &&
                         ((total & 3) == 0) && ((base_ev & 3) == 0);

    // ---- Stage this block's event indices into LDS ----------------------
    if (staged) {
        const int nvec = total >> 2;                 // int4 chunks
#if HAVE_ASYNC_LDS
        for (int t = tid; t < nvec; t += blockDim.x) {
            __builtin_amdgcn_global_load_async_to_lds_b128(
                (as1_v4i*)(ev + base_ev + t * 4),
                (as3_v4i*)(&s_idx[t * 4]),
                /*offset=*/0, /*cpol=*/0);
        }
        __builtin_amdgcn_s_wait_asynccnt(0);
#else
        const int4* evv = reinterpret_cast<const int4*>(ev + base_ev);
        for (int t = tid; t < nvec; t += blockDim.x) {
            reinterpret_cast<int4*>(s_idx)[t] = evv[t];
        }
#endif
    }
    __syncthreads();

    // ---- Half-wave per user: 16 lanes x float4 = one 256B row per load --
    const int lane = tid & 31;          // wave32
    const int wave = tid >> 5;
    const int half = lane >> 4;         // 0 or 1: which user of this wave
    const int sub  = lane & 15;         // float4 slot within the row (64 dims)

    const int u = u0 + wave * 2 + half;
    if (u >= n_users) return;

    const int start = offs[u];
    const int end   = offs[u + 1];

    const float4* __restrict__ emb4 = reinterpret_cast<const float4*>(emb);

    float ax = 0.f, ay = 0.f, az = 0.f, aw = 0.f;

    if (staged) {
        int r = start - base_ev;
        const int rend = end - base_ev;
        // 4x unrolled: 4 independent b128 gathers in flight per iteration
        for (; r + 4 <= rend; r += 4) {
            const int i0 = s_idx[r + 0];
            const int i1 = s_idx[r + 1];
            const int i2 = s_idx[r + 2];
            const int i3 = s_idx[r + 3];
            const float4 v0 = emb4[(uint32_t)i0 * 16u + (uint32_t)sub];
            const float4 v1 = emb4[(uint32_t)i1 * 16u + (uint32_t)sub];
            const float4 v2 = emb4[(uint32_t)i2 * 16u + (uint32_t)sub];
            const float4 v3 = emb4[(uint32_t)i3 * 16u + (uint32_t)sub];
            if (r + 8 < rend) {  // prefetch 2 unroll-blocks ahead -> global_prefetch_b8
                __builtin_prefetch(&emb4[(uint32_t)s_idx[r + 8] * 16u + (uint32_t)sub], 0, 0);
            }
            ax += v0.x; ay += v0.y; az += v0.z; aw += v0.w;
            ax += v1.x; ay += v1.y; az += v1.z; aw += v1.w;
            ax += v2.x; ay += v2.y; az += v2.z; aw += v2.w;
            ax += v3.x; ay += v3.y; az += v3.z; aw += v3.w;
        }
        for (; r < rend; ++r) {
            const float4 v = emb4[(uint32_t)s_idx[r] * 16u + (uint32_t)sub];
            ax += v.x; ay += v.y; az += v.z; aw += v.w;
        }
    } else {
        // Fallback (irregular offsets): gather indices straight from global
        for (int e = start; e < end; ++e) {
            const float4 v = emb4[(uint32_t)ev[e] * 16u + (uint32_t)sub];
            ax += v.x; ay += v.y; az += v.z; aw += v.w;
        }
    }

    int len = end - start;
    if (len < 1) len = 1;
    const float den = (float)len;

    float4 o;
    o.x = ax / den; o.y = ay / den; o.z = az / den; o.w = aw / den;
    reinterpret_cast<float4*>(out)[(uint32_t)u * 16u + (uint32_t)sub] = o;
}

extern "C" void kernel_launch(void* const* d_in, const int* in_sizes, int n_in,
                              void* d_out, int out_size, void* d_ws, size_t ws_size,
                              hipStream_t stream) {
    (void)n_in; (void)out_size; (void)d_ws; (void)ws_size;
    const int*   ev   = (const int*)d_in[0];     // event_indices [TOTAL] int32
    const int*   offs = (const int*)d_in[1];     // offsets [n_users+1] int32
    const float* emb  = (const float*)d_in[2];   // emb_weight [N_SAMPLES, 64] fp32
    float*       out  = (float*)d_out;           // [n_users, 64] fp32

    const int n_users = in_sizes[1] - 1;
    const int blocks  = (n_users + BLOCK_USERS - 1) / BLOCK_USERS;

    VariableSortedHistoryPooling_kernel<<<blocks, 256, 0, stream>>>(
        ev, offs, emb, out, n_users);
}